// EquivariantLayer_34634616275601
// MI455X (gfx1250) — compile-verified
//
#include <hip/hip_runtime.h>
#include <hip/hip_bf16.h>
#include <math.h>

#define HIDDEN   32
#define HEADS    4
#define PROJ     128
#define LN_EPS   1e-5f
#define ACC_STRIDE 17           // per node: 4 heads x (3 rel + 1 attn) + count
#define INV_SQRT32 0.17677669529663687f

typedef __attribute__((ext_vector_type(2))) float v2f;
typedef __attribute__((ext_vector_type(8))) float v8f;

// ---------------------------------------------------------------------------
// Kernel 0: zero the per-node accumulators in workspace
// ---------------------------------------------------------------------------
__global__ void zero_acc_kernel(float* acc, long long total) {
    long long i = (long long)blockIdx.x * blockDim.x + threadIdx.x;
    if (i < total) acc[i] = 0.0f;
}

// ---------------------------------------------------------------------------
// Kernel 1: precompute
//   qkc[h*13 + 0..8]  = G_h (3x3)   = (Wq_h Wk_h^T) / sqrt(32)
//   qkc[h*13 + 9..11] = g_h (3)     = (Wq_h bk_h + Wk_h bq_h) / sqrt(32)
//   qkc[h*13 + 12]    = c_h         = (bq_h . bk_h) / sqrt(32)
//   Bm[r*32 + o], r = h*4 + i:  i<3 -> sum_d Wv[i, h*32+d] * Wout[h*32+d, o]
//                               i=3 -> sum_d bv[h*32+d]    * Wout[h*32+d, o]
// ---------------------------------------------------------------------------
__global__ void precompute_kernel(const float* __restrict__ Wq, const float* __restrict__ bq,
                                  const float* __restrict__ Wk, const float* __restrict__ bk,
                                  const float* __restrict__ Wv, const float* __restrict__ bv,
                                  const float* __restrict__ Wout,
                                  float* __restrict__ qkc, float* __restrict__ Bm) {
    int t = threadIdx.x;
    if (t < 64) {
        int h = t >> 4, sub = t & 15;
        int base = h * HIDDEN;                  // column offset into PROJ
        if (sub < 9) {                          // G_h[i][j]
            int i = sub / 3, j = sub % 3;
            float s = 0.0f;
            for (int d = 0; d < HIDDEN; ++d)
                s += Wq[i * PROJ + base + d] * Wk[j * PROJ + base + d];
            qkc[h * 13 + sub] = s * INV_SQRT32;
        } else if (sub < 12) {                  // g_h[i]
            int i = sub - 9;
            float s = 0.0f;
            for (int d = 0; d < HIDDEN; ++d)
                s += Wq[i * PROJ + base + d] * bk[base + d]
                   + Wk[i * PROJ + base + d] * bq[base + d];
            qkc[h * 13 + sub] = s * INV_SQRT32;
        } else if (sub == 12) {                 // c_h
            float s = 0.0f;
            for (int d = 0; d < HIDDEN; ++d)
                s += bq[base + d] * bk[base + d];
            qkc[h * 13 + 12] = s * INV_SQRT32;
        }
    } else {
        int t2 = t - 64;                        // 0..511
        int r = t2 >> 5, o = t2 & 31;
        int h = r >> 2, i = r & 3;
        int base = h * HIDDEN;
        float s = 0.0f;
        if (i < 3) {
            for (int d = 0; d < HIDDEN; ++d)
                s += Wv[i * PROJ + base + d] * Wout[(base + d) * HIDDEN + o];
        } else {
            for (int d = 0; d < HIDDEN; ++d)
                s += bv[base + d] * Wout[(base + d) * HIDDEN + o];
        }
        Bm[r * HIDDEN + o] = s;
    }
}

// ---------------------------------------------------------------------------
// Kernel 2: per-edge — quadratic-form attention scores, softmax over heads,
// scatter (attn_h * rel, attn_h, 1) into node accumulators (17 f32 atomics).
// ---------------------------------------------------------------------------
__global__ void edge_kernel(const float* __restrict__ pos,
                            const long long* __restrict__ ei, int nEdges,
                            const float* __restrict__ qkc,
                            float* __restrict__ acc) {
    int e = blockIdx.x * blockDim.x + threadIdx.x;
    if (e >= nEdges) return;
    long long r = ei[e];
    long long c = ei[(long long)nEdges + e];
    float rx = pos[r * 3 + 0] - pos[c * 3 + 0];
    float ry = pos[r * 3 + 1] - pos[c * 3 + 1];
    float rz = pos[r * 3 + 2] - pos[c * 3 + 2];

    float s[HEADS];
#pragma unroll
    for (int h = 0; h < HEADS; ++h) {
        const float* q = qkc + h * 13;
        s[h] = rx * (q[0] * rx + q[1] * ry + q[2] * rz)
             + ry * (q[3] * rx + q[4] * ry + q[5] * rz)
             + rz * (q[6] * rx + q[7] * ry + q[8] * rz)
             + q[9] * rx + q[10] * ry + q[11] * rz + q[12];
    }
    float m = fmaxf(fmaxf(s[0], s[1]), fmaxf(s[2], s[3]));
    float ex[HEADS], sum = 0.0f;
#pragma unroll
    for (int h = 0; h < HEADS; ++h) { ex[h] = __expf(s[h] - m); sum += ex[h]; }
    float inv = 1.0f / sum;

    float* dst = acc + (size_t)c * ACC_STRIDE;
#pragma unroll
    for (int h = 0; h < HEADS; ++h) {
        float a = ex[h] * inv;
        atomicAdd(dst + h * 4 + 0, a * rx);
        atomicAdd(dst + h * 4 + 1, a * ry);
        atomicAdd(dst + h * 4 + 2, a * rz);
        atomicAdd(dst + h * 4 + 3, a);
    }
    atomicAdd(dst + 16, 1.0f);
}

// ---------------------------------------------------------------------------
// Kernel 3: per-node — WMMA [16 nodes x 16] @ [16 x 32] fused matmul
// (V_WMMA_F32_16X16X4_F32, 4 K-steps x 2 N-tiles), then LayerNorm + SiLU.
// Each wave owns 16 nodes; block = 256 threads = 8 waves = 128 nodes.
// ---------------------------------------------------------------------------
__global__ void __launch_bounds__(256)
node_kernel(const float* __restrict__ acc, const float* __restrict__ Bm,
            const float* __restrict__ bout, const float* __restrict__ gamma,
            const float* __restrict__ beta, float* __restrict__ out, int nNodes) {
    __shared__ float lds[8][16][32];            // per-wave 16x32 f32 tile

    const int lane = threadIdx.x & 31;
    const int wib  = threadIdx.x >> 5;          // wave in block (0..7)
    const int waveId = blockIdx.x * 8 + wib;
    const int nodeBase = waveId * 16;

    const int m    = lane & 15;                 // M row / N column (per ISA layout)
    const int kgrp = lane >> 4;                 // lanes 16-31 take K+2/K+3 (A), rows M+8 (C/D)

    int node = nodeBase + m;
    int nclamp = node < nNodes ? node : 0;      // keep EXEC all-ones through WMMA
    const float* arow = acc + (size_t)nclamp * ACC_STRIDE;
    float cnt  = arow[16];
    float invc = 1.0f / fmaxf(cnt, 1.0f);

    v8f c0 = {}; v8f c1 = {};
#pragma unroll
    for (int step = 0; step < 4; ++step) {
        int k0 = step * 4 + kgrp * 2;
        v2f a;  a.x  = arow[k0] * invc;           a.y  = arow[k0 + 1] * invc;
        v2f b0; b0.x = Bm[k0 * 32 + m];           b0.y = Bm[(k0 + 1) * 32 + m];
        v2f b1; b1.x = Bm[k0 * 32 + 16 + m];      b1.y = Bm[(k0 + 1) * 32 + 16 + m];
        c0 = __builtin_amdgcn_wmma_f32_16x16x4_f32(false, a, false, b0, (short)0, c0, false, false);
        c1 = __builtin_amdgcn_wmma_f32_16x16x4_f32(false, a, false, b1, (short)0, c1, false, false);
    }

    // C/D layout: VGPR r -> (M=r, N=lane) for lanes 0-15, (M=r+8, N=lane-16) for 16-31
    float* my = &lds[wib][0][0];
    const int row0 = kgrp ? 8 : 0;
#pragma unroll
    for (int r = 0; r < 8; ++r) {
        my[(row0 + r) * 32 + m]      = c0[r];
        my[(row0 + r) * 32 + 16 + m] = c1[r];
    }
    __syncthreads();

    // LayerNorm + SiLU: lane pair (L, L^16) handles node m, 16 features each
    float vbuf[16];
    float s = 0.0f, sq = 0.0f;
#pragma unroll
    for (int j = 0; j < 16; ++j) {
        int o = kgrp * 16 + j;
        float v = my[m * 32 + o] + bout[o];
        vbuf[j] = v; s += v; sq += v * v;
    }
    s  += __shfl_xor(s, 16, 32);
    sq += __shfl_xor(sq, 16, 32);
    float mean = s * (1.0f / 32.0f);
    float var  = sq * (1.0f / 32.0f) - mean * mean;
    float rstd = rsqrtf(var + LN_EPS);

    if (node < nNodes) {
#pragma unroll
        for (int j = 0; j < 16; ++j) {
            int o = kgrp * 16 + j;
            float t = (vbuf[j] - mean) * rstd * gamma[o] + beta[o];
            out[(size_t)node * 32 + o] = t / (1.0f + __expf(-t));   // SiLU
        }
    }
}

// ---------------------------------------------------------------------------
extern "C" void kernel_launch(void* const* d_in, const int* in_sizes, int n_in,
                              void* d_out, int out_size, void* d_ws, size_t ws_size,
                              hipStream_t stream) {
    const float*     pos   = (const float*)d_in[0];
    const long long* ei    = (const long long*)d_in[1];   // edge_index int64 [2,E]
    const float*     Wq    = (const float*)d_in[2];
    const float*     bq    = (const float*)d_in[3];
    const float*     Wk    = (const float*)d_in[4];
    const float*     bk    = (const float*)d_in[5];
    const float*     Wv    = (const float*)d_in[6];
    const float*     bv    = (const float*)d_in[7];
    const float*     Wout  = (const float*)d_in[8];
    const float*     bout  = (const float*)d_in[9];
    const float*     gamma = (const float*)d_in[10];
    const float*     beta  = (const float*)d_in[11];
    float*           out   = (float*)d_out;

    const int nNodes = in_sizes[0] / 3;
    const int nEdges = in_sizes[1] / 2;

    // workspace layout (floats): [acc: nNodes*17][qkc: 64][Bm: 512]
    float* acc = (float*)d_ws;
    float* qkc = acc + (size_t)nNodes * ACC_STRIDE;
    float* Bm  = qkc + 64;

    long long accTotal = (long long)nNodes * ACC_STRIDE;
    int zblocks = (int)((accTotal + 255) / 256);
    zero_acc_kernel<<<zblocks, 256, 0, stream>>>(acc, accTotal);

    precompute_kernel<<<1, 576, 0, stream>>>(Wq, bq, Wk, bk, Wv, bv, Wout, qkc, Bm);

    edge_kernel<<<(nEdges + 255) / 256, 256, 0, stream>>>(pos, ei, nEdges, qkc, acc);

    int nblocks = (nNodes + 127) / 128;        // 128 nodes per 256-thread block
    node_kernel<<<nblocks, 256, 0, stream>>>(acc, Bm, bout, gamma, beta, out, nNodes);
}